// GNN_35459249996470
// MI455X (gfx1250) — compile-verified
//
#include <hip/hip_runtime.h>
#include <hip/hip_bf16.h>
#include <stdint.h>

// ---------------------------------------------------------------------------
// GraphSAGE 2-layer forward for MI455X (gfx1250, wave32, WMMA).
//   N = 100000 nodes, E = 1.6M edges, D = 128 everywhere.
//   layer(x) = mean_agg(x) @ W_l^T + b_l + x @ W_r^T   (ReLU after layer 1)
// Strategy:
//   - edge aggregation: wave-per-edge f32 atomic scatter (L2-resident, 192MB L2)
//   - dense part: fused K=256 GEMM per layer (A = [mean | x], B = [W_l^T;W_r^T])
//     via V_WMMA_F32_16X16X4_F32 (fp32-exact vs fp32 reference).
//   - A tile: global_load_async_to_lds_b128 (CDNA5 async copy, ASYNCcnt).
//   - B: pre-packed in WMMA per-lane register layout -> 4 coalesced
//     global_load_b128 per K-chunk straight into WMMA operands (no LDS).
// ---------------------------------------------------------------------------

typedef __attribute__((ext_vector_type(2))) float v2f;
typedef __attribute__((ext_vector_type(8))) float v8f;

#define D_FEAT 128
#define K_TOT  256          // [mean | x]
#define LSA_STRIDE 260      // pad: 16B-aligned rows, banks (4*lr+kA)%64 distinct

// -------------------------------- zero fill --------------------------------
__global__ void gnn_zero_kernel(float* __restrict__ p, long long n) {
  long long i = (long long)blockIdx.x * blockDim.x + threadIdx.x;
  if (i < n) p[i] = 0.0f;
}

// ---- pack B = [W_l^T ; W_r^T] into WMMA per-lane layout -------------------
// Flat index t = ((((kc*8 + w)*4 + q)*32 + lane)*4 + j, so the GEMM kernel's
// per-chunk fetch "float4 slot q of lane `lane` for wave w" is contiguous
// across lanes -> perfectly coalesced 512B global_load_b128 per wave.
//   kk = q*2 + (j>>1), c = j&1, k = kc*32 + kk*4 + (lane>>4)*2 + c,
//   col = w*16 + (lane&15);  B[k][col] = (k<128 ? Wl[col][k] : Wr[col][k-128])
__global__ void gnn_wt_kernel(const float* __restrict__ Wl,
                              const float* __restrict__ Wr,
                              float* __restrict__ WTp) {
  int t = blockIdx.x * blockDim.x + threadIdx.x;   // 32768 elements
  if (t >= K_TOT * D_FEAT) return;
  int j    = t & 3;
  int lane = (t >> 2) & 31;
  int q    = (t >> 7) & 3;
  int w    = (t >> 9) & 7;
  int kc   = t >> 12;
  int kk   = q * 2 + (j >> 1);
  int c    = j & 1;
  int half = lane >> 4;
  int lr   = lane & 15;
  int k    = kc * 32 + kk * 4 + half * 2 + c;
  int col  = w * 16 + lr;
  WTp[t] = (k < D_FEAT) ? Wl[col * D_FEAT + k]
                        : Wr[col * D_FEAT + (k - D_FEAT)];
}

// ------------------------------ degree count -------------------------------
__global__ void gnn_degree_kernel(const int* __restrict__ dst,
                                  float* __restrict__ deg, int nEdges) {
  int e = blockIdx.x * blockDim.x + threadIdx.x;
  if (e < nEdges) atomicAdd(&deg[dst[e]], 1.0f);
}

// ---------------------- edge gather + atomic scatter -----------------------
// one wave32 per edge; each lane moves a float4 (whole 512B row per wave)
__global__ void gnn_scatter_kernel(const float* __restrict__ feat,
                                   const int* __restrict__ src,
                                   const int* __restrict__ dst,
                                   float* __restrict__ agg, int nEdges) {
  long long t = (long long)blockIdx.x * blockDim.x + threadIdx.x;
  int e    = (int)(t >> 5);
  int lane = (int)(t & 31);
  if (e >= nEdges) return;
  int s = src[e];
  int d = dst[e];
  const float4 v = *(const float4*)(feat + (long long)s * D_FEAT + lane * 4);
  float* o = agg + (long long)d * D_FEAT + lane * 4;
  atomicAdd(o + 0, v.x);
  atomicAdd(o + 1, v.y);
  atomicAdd(o + 2, v.z);
  atomicAdd(o + 3, v.w);
}

// ------------------- fused SAGE GEMM:  out = A @ B + bias ------------------
//   A[r] = [ agg[r] (scaled by 1/deg at consume time) | xin[r] ]  (16 x 256)
//   B packed as WTp (see gnn_wt_kernel).
// block = 256 threads = 8 wave32; wave w owns the 16x16 tile at col w*16.
// 64 x v_wmma_f32_16x16x4_f32 per wave.
template <bool RELU>
__global__ __launch_bounds__(256)
void gnn_sage_gemm_kernel(const float* __restrict__ agg,
                          const float* __restrict__ deg,
                          const float* __restrict__ xin,
                          const float4* __restrict__ WTp,
                          const float* __restrict__ bias,
                          float* __restrict__ out, int nNodes) {
  __shared__ float lsA[16 * LSA_STRIDE];

  const int t       = threadIdx.x;
  const int rowBase = blockIdx.x * 16;

  // ---- async-copy A tile (16 rows x 256 floats) into LDS, 16B per lane ----
  // granule f = t + i*256 : row = f>>6, c4 = f&63 (float4 column index)
#pragma unroll
  for (int i = 0; i < 4; ++i) {
    int f    = t + i * 256;
    int row  = f >> 6;
    int c4   = f & 63;
    int node = rowBase + row;
    if (node < nNodes) {
      const float* gsrc = (c4 < 32)
          ? (agg + (long long)node * D_FEAT + c4 * 4)
          : (xin + (long long)node * D_FEAT + (c4 - 32) * 4);
      uint32_t laddr = (uint32_t)(uintptr_t)(&lsA[row * LSA_STRIDE + c4 * 4]);
      asm volatile("global_load_async_to_lds_b128 %0, %1, off"
                   :: "v"(laddr), "v"(gsrc) : "memory");
    }
  }
  asm volatile("s_wait_asynccnt 0" ::: "memory");
  __syncthreads();

  const int wave    = t >> 5;          // 0..7
  const int lane    = t & 31;
  const int half    = lane >> 4;       // K-sub-pair select
  const int lr      = lane & 15;       // row (A) / col (B,C,D) within tile
  const int colBase = wave * 16;

  // per-lane 1/deg for this lane's A row (folded mean scaling)
  float invRow = 1.0f;
  {
    int nodeR = rowBase + lr;
    if (nodeR < nNodes) invRow = 1.0f / fmaxf(deg[nodeR], 1.0f);
  }

  v8f acc = {};                        // fp32 16x16 accumulator (8 VGPRs)

  for (int kc = 0; kc < 8; ++kc) {     // K chunks of 32
    // B operands for this chunk: 4 coalesced b128 loads, already in WMMA layout
    float4 bq[4];
#pragma unroll
    for (int q = 0; q < 4; ++q)
      bq[q] = WTp[((kc * 8 + wave) * 4 + q) * 32 + lane];

    const float aScale = (kc < 4) ? invRow : 1.0f;   // mean half vs x half

#pragma unroll
    for (int kk = 0; kk < 8; ++kk) {   // 8 x K=4 WMMA steps
      const int kA = kc * 32 + kk * 4 + half * 2;
      v2f a, b;
      // A 16x4 layout: lanes 0-15 K={0,1}, lanes 16-31 K={2,3}, M = lr
      a.x = lsA[lr * LSA_STRIDE + kA] * aScale;
      a.y = lsA[lr * LSA_STRIDE + kA + 1] * aScale;
      const float4 q4 = bq[kk >> 1];
      if (kk & 1) { b.x = q4.z; b.y = q4.w; }
      else        { b.x = q4.x; b.y = q4.y; }
      acc = __builtin_amdgcn_wmma_f32_16x16x4_f32(
          /*neg_a=*/false, a, /*neg_b=*/false, b,
          /*c_mod=*/(short)0, acc, /*reuse_a=*/false, /*reuse_b=*/false);
    }
  }

  // ---- epilogue: C/D layout -> VGPR j, lanes 0-15: M=j, lanes 16-31: M=j+8
#pragma unroll
  for (int j = 0; j < 8; ++j) {
    int row = rowBase + j + half * 8;
    int col = colBase + lr;
    if (row < nNodes) {
      float v = acc[j] + bias[col];
      if (RELU) v = fmaxf(v, 0.0f);
      out[(long long)row * D_FEAT + col] = v;
    }
  }
}

// ---------------------------------------------------------------------------
extern "C" void kernel_launch(void* const* d_in, const int* in_sizes, int n_in,
                              void* d_out, int out_size, void* d_ws, size_t ws_size,
                              hipStream_t stream) {
  const float* x    = (const float*)d_in[0];
  const int*   edge = (const int*)d_in[1];
  const float* W1l  = (const float*)d_in[2];
  const float* b1l  = (const float*)d_in[3];
  const float* W1r  = (const float*)d_in[4];
  const float* W2l  = (const float*)d_in[5];
  const float* b2l  = (const float*)d_in[6];
  const float* W2r  = (const float*)d_in[7];
  float* out = (float*)d_out;

  const int nNodes = in_sizes[0] / D_FEAT;     // 100000
  const int nEdges = in_sizes[1] / 2;          // 1600000
  const int* src = edge;
  const int* dst = edge + nEdges;

  // workspace layout (floats): deg | agg | WT1p | WT2p
  float* ws   = (float*)d_ws;
  float* deg  = ws;
  float* agg  = deg + nNodes;
  float* WT1  = agg + (long long)nNodes * D_FEAT;   // byte off 51.6e6, 16B-aligned
  float* WT2  = WT1 + K_TOT * D_FEAT;

  const int TB = 256;
  const long long aggN = (long long)nNodes * D_FEAT;
  const int gridRows = (nNodes + 15) / 16;      // 6250
  const long long scatT = (long long)nEdges * 32;

  // ---- setup: zero deg+agg, pack transposed weights, degrees
  gnn_zero_kernel<<<(int)((nNodes + TB - 1) / TB), TB, 0, stream>>>(deg, nNodes);
  gnn_zero_kernel<<<(int)((aggN + TB - 1) / TB), TB, 0, stream>>>(agg, aggN);
  gnn_wt_kernel<<<(K_TOT * D_FEAT + TB - 1) / TB, TB, 0, stream>>>(W1l, W1r, WT1);
  gnn_wt_kernel<<<(K_TOT * D_FEAT + TB - 1) / TB, TB, 0, stream>>>(W2l, W2r, WT2);
  gnn_degree_kernel<<<(nEdges + TB - 1) / TB, TB, 0, stream>>>(dst, deg, nEdges);

  // ---- layer 1: scatter x, fused GEMM (+ReLU) -> d_out (= h)
  gnn_scatter_kernel<<<(int)((scatT + TB - 1) / TB), TB, 0, stream>>>(
      x, src, dst, agg, nEdges);
  gnn_sage_gemm_kernel<true><<<gridRows, TB, 0, stream>>>(
      agg, deg, x, (const float4*)WT1, b1l, out, nNodes);

  // ---- layer 2: re-zero agg, scatter h, fused GEMM -> d_out
  gnn_zero_kernel<<<(int)((aggN + TB - 1) / TB), TB, 0, stream>>>(agg, aggN);
  gnn_scatter_kernel<<<(int)((scatT + TB - 1) / TB), TB, 0, stream>>>(
      out, src, dst, agg, nEdges);
  gnn_sage_gemm_kernel<false><<<gridRows, TB, 0, stream>>>(
      agg, deg, out, (const float4*)WT2, b2l, out, nNodes);
}